// StructureGNN_49168785604678
// MI455X (gfx1250) — compile-verified
//
#include <hip/hip_runtime.h>
#include <hip/hip_bf16.h>

// ---------------- problem constants (match reference) ----------------
constexpr int Nn   = 50000;   // nodes
constexpr int Ee   = 640000;  // edges
constexpr int Gg   = 64;      // graphs
constexpr int F_IN = 24;
constexpr int Hd   = 128;     // hidden
constexpr int Ll   = 4;       // layers
constexpr float BN_EPS = 1e-5f;

typedef float v2f __attribute__((ext_vector_type(2)));
typedef float v8f __attribute__((ext_vector_type(8)));

__device__ __forceinline__ void atomAddF(float* p, float v) {
  unsafeAtomicAdd(p, v);   // native global_atomic_add_f32 on CDNA
}

// ---------------- utility fills ----------------
__global__ void k_fill(float* __restrict__ p, int n, float v) {
  int i = blockIdx.x * blockDim.x + threadIdx.x;
  if (i < n) p[i] = v;
}

// ---------------- degree / normalization ----------------
__global__ void k_deg(const int* __restrict__ dst, float* __restrict__ deg, int e) {
  int i = blockIdx.x * blockDim.x + threadIdx.x;
  if (i < e) atomAddF(&deg[dst[i]], 1.0f);
}
__global__ void k_rsqrt(float* __restrict__ p, int n) {
  int i = blockIdx.x * blockDim.x + threadIdx.x;
  if (i < n) p[i] = rsqrtf(p[i]);
}

// ---------------- WMMA fp32 GEMM:  C[M x 128] = A[M x K] @ W[K x 128] ----------------
// block: 256 threads = 8 waves; block tile 64(M) x 128(N); wave tile 16(M) x 64(N).
// Uses V_WMMA_F32_16X16X4_F32; K must be a multiple of 4 (24 and 128 both are).
__global__ void k_gemm_wmma(const float* __restrict__ A, const float* __restrict__ W,
                            float* __restrict__ C, int M, int K) {
  const int tid  = threadIdx.x;
  const int wave = tid >> 5;
  const int lane = tid & 31;
  const int half = lane >> 4;     // 0: lanes 0-15, 1: lanes 16-31
  const int l16  = lane & 15;
  const int wy   = wave >> 1;     // 0..3  (M sub-tile)
  const int wx   = wave & 1;      // 0..1  (N half)
  const int m0   = blockIdx.x * 64 + wy * 16;
  const int n0   = wx * 64;

  v8f acc0 = {}, acc1 = {}, acc2 = {}, acc3 = {};

  // A-fragment row for this lane (clamp OOB rows; stores are masked)
  int arow = m0 + l16;
  if (arow >= M) arow = M - 1;
  const float* __restrict__ Arow = A + (size_t)arow * K;

  for (int kk = 0; kk < K; kk += 4) {
    // 32-bit A 16x4 layout: v0 = K=kk (lo lanes) / kk+2 (hi lanes); v1 = kk+1 / kk+3
    v2f a;
    a.x = Arow[kk + 2 * half + 0];
    a.y = Arow[kk + 2 * half + 1];

    const float* __restrict__ Wr0 = W + (size_t)(kk + 2 * half) * Hd;
    const float* __restrict__ Wr1 = Wr0 + Hd;
    v2f b0, b1, b2, b3;
    b0.x = Wr0[n0      + l16]; b0.y = Wr1[n0      + l16];
    b1.x = Wr0[n0 + 16 + l16]; b1.y = Wr1[n0 + 16 + l16];
    b2.x = Wr0[n0 + 32 + l16]; b2.y = Wr1[n0 + 32 + l16];
    b3.x = Wr0[n0 + 48 + l16]; b3.y = Wr1[n0 + 48 + l16];

    acc0 = __builtin_amdgcn_wmma_f32_16x16x4_f32(false, a, false, b0, (short)0, acc0, false, false);
    acc1 = __builtin_amdgcn_wmma_f32_16x16x4_f32(false, a, false, b1, (short)0, acc1, false, false);
    acc2 = __builtin_amdgcn_wmma_f32_16x16x4_f32(false, a, false, b2, (short)0, acc2, false, false);
    acc3 = __builtin_amdgcn_wmma_f32_16x16x4_f32(false, a, false, b3, (short)0, acc3, false, false);
  }

  // C layout: VGPR r holds row (r + 8*half), col = l16
  #pragma unroll
  for (int r = 0; r < 8; ++r) {
    int row = m0 + r + 8 * half;
    if (row < M) {
      float* __restrict__ Crow = C + (size_t)row * Hd + n0 + l16;
      Crow[0]  = acc0[r];
      Crow[16] = acc1[r];
      Crow[32] = acc2[r];
      Crow[48] = acc3[r];
    }
  }
}

// ---------------- edge scatter: agg[dst] += hW[src] * dis[src]*dis[dst] ----------------
// one wave per edge; 4 floats per lane (128 features)
__global__ void k_scatter(const float* __restrict__ hW, const int* __restrict__ src,
                          const int* __restrict__ dst, const float* __restrict__ dis,
                          float* __restrict__ agg, int e) {
  int lane = threadIdx.x & 31;
  int edge = (blockIdx.x * blockDim.x + threadIdx.x) >> 5;
  if (edge >= e) return;
  int s = src[edge];
  int d = dst[edge];
  float c = dis[s] * dis[d];
  const float4* __restrict__ in = (const float4*)(hW + (size_t)s * Hd) + lane;
  float4 v = *in;
  float* __restrict__ outp = agg + (size_t)d * Hd + lane * 4;
  atomAddF(outp + 0, v.x * c);
  atomAddF(outp + 1, v.y * c);
  atomAddF(outp + 2, v.z * c);
  atomAddF(outp + 3, v.w * c);
}

// ---------------- self-loop + bias + BN(eval) + ReLU, h written in place ----------------
__global__ void k_finalize(const float* __restrict__ hW, float* __restrict__ agg,
                           const float* __restrict__ dis,
                           const float* __restrict__ bias, const float* __restrict__ gamma,
                           const float* __restrict__ beta, const float* __restrict__ mean,
                           const float* __restrict__ var, int total) {
  int idx = blockIdx.x * blockDim.x + threadIdx.x;
  if (idx >= total) return;
  int i = idx >> 7;          // node
  int f = idx & (Hd - 1);    // feature
  float di = dis[i];
  float v = agg[idx] + hW[idx] * di * di + bias[f];
  v = (v - mean[f]) * rsqrtf(var[f] + BN_EPS) * gamma[f] + beta[f];
  agg[idx] = fmaxf(v, 0.0f);
}

// ---------------- pooling: exploits sorted batch ids, run-length local accumulation ----
constexpr int POOL_CHUNK = 256;
__global__ void k_pool(const float* __restrict__ h, const int* __restrict__ batch,
                       float* __restrict__ gsum, float* __restrict__ gmax,
                       float* __restrict__ gcnt) {
  int f  = threadIdx.x;                 // 0..127
  int i0 = blockIdx.x * POOL_CHUNK;
  int iend = i0 + POOL_CHUNK; if (iend > Nn) iend = Nn;
  int curg = batch[i0];
  float s = 0.f, m = 0.f, cnt = 0.f;    // h >= 0 post-ReLU, so 0 is a safe max identity
  for (int i = i0; i < iend; ++i) {
    int g = batch[i];
    if (g != curg) {
      atomAddF(&gsum[curg * Hd + f], s);
      atomicMax((int*)&gmax[curg * Hd + f], __float_as_int(m));
      if (f == 0) atomAddF(&gcnt[curg], cnt);
      s = 0.f; m = 0.f; cnt = 0.f; curg = g;
    }
    float v = h[(size_t)i * Hd + f];
    s += v; m = fmaxf(m, v); cnt += 1.f;
  }
  atomAddF(&gsum[curg * Hd + f], s);
  atomicMax((int*)&gmax[curg * Hd + f], __float_as_int(m));
  if (f == 0) atomAddF(&gcnt[curg], cnt);
}

// ---------------- tiny MLP head (G=64, negligible FLOPs) ----------------
__global__ void k_mlp1(const float* __restrict__ gsum, const float* __restrict__ gmax,
                       const float* __restrict__ gcnt, const float* __restrict__ pW,
                       const float* __restrict__ pb, float* __restrict__ hid) {
  int g = blockIdx.x;      // 0..G-1
  int j = threadIdx.x;     // 0..H-1
  float inv = 1.0f / fmaxf(gcnt[g], 1.0f);
  float acc = pb[j];
  #pragma unroll 4
  for (int k = 0; k < Hd; ++k)
    acc += (gsum[g * Hd + k] * inv) * pW[k * Hd + j];
  #pragma unroll 4
  for (int k = 0; k < Hd; ++k)
    acc += gmax[g * Hd + k] * pW[(Hd + k) * Hd + j];
  hid[g * Hd + j] = fmaxf(acc, 0.0f);
}
__global__ void k_mlp2(const float* __restrict__ hid, const float* __restrict__ oW,
                       const float* __restrict__ ob, float* __restrict__ out) {
  int g = blockIdx.x;      // 0..G-1
  int j = threadIdx.x;     // 0..H/2-1
  float acc = ob[j];
  #pragma unroll 4
  for (int k = 0; k < Hd; ++k)
    acc += hid[g * Hd + k] * oW[k * (Hd / 2) + j];
  out[g * (Hd / 2) + j] = acc;
}

// ---------------- launcher ----------------
extern "C" void kernel_launch(void* const* d_in, const int* in_sizes, int n_in,
                              void* d_out, int out_size, void* d_ws, size_t ws_size,
                              hipStream_t stream) {
  const float* x      = (const float*)d_in[0];
  const int*   ei     = (const int*)d_in[1];   // [2, E]: first E = src, next E = dst
  const int*   batch  = (const int*)d_in[2];
  const float* W0     = (const float*)d_in[3];
  const float* Wh     = (const float*)d_in[4];
  const float* b      = (const float*)d_in[5];
  const float* gamma  = (const float*)d_in[6];
  const float* beta   = (const float*)d_in[7];
  const float* mean   = (const float*)d_in[8];
  const float* var    = (const float*)d_in[9];
  const float* poolW  = (const float*)d_in[10];
  const float* poolB  = (const float*)d_in[11];
  const float* outW   = (const float*)d_in[12];
  const float* outB   = (const float*)d_in[13];
  float*       out    = (float*)d_out;

  const int* srcp = ei;
  const int* dstp = ei + Ee;

  // workspace layout (floats): ~51.5 MB total
  float* ws   = (float*)d_ws;
  float* bufA = ws;                               // N*H  (h / agg, ping)
  float* bufB = bufA + (size_t)Nn * Hd;           // N*H  (hW, pong)
  float* dis  = bufB + (size_t)Nn * Hd;           // N
  float* gsum = dis + Nn;                         // G*H
  float* gmax = gsum + Gg * Hd;                   // G*H
  float* gcnt = gmax + Gg * Hd;                   // G
  float* hid  = gcnt + Gg;                        // G*H

  const int TPB = 256;

  // 1) symmetric normalization coefficients: dis = rsqrt(deg), deg includes self-loop
  k_fill<<<(Nn + TPB - 1) / TPB, TPB, 0, stream>>>(dis, Nn, 1.0f);
  k_deg<<<(Ee + TPB - 1) / TPB, TPB, 0, stream>>>(dstp, dis, Ee);
  k_rsqrt<<<(Nn + TPB - 1) / TPB, TPB, 0, stream>>>(dis, Nn);

  // 2) GCN layers
  const float* hin = x;
  int K = F_IN;
  const int total = Nn * Hd;
  for (int l = 0; l < Ll; ++l) {
    const float* Wl = (l == 0) ? W0 : (Wh + (size_t)(l - 1) * Hd * Hd);
    k_gemm_wmma<<<(Nn + 63) / 64, 256, 0, stream>>>(hin, Wl, bufB, Nn, K);
    k_fill<<<(total + TPB - 1) / TPB, TPB, 0, stream>>>(bufA, total, 0.0f);
    // one wave per edge -> 8 edges per 256-thread block
    k_scatter<<<(Ee + 7) / 8, 256, 0, stream>>>(bufB, srcp, dstp, dis, bufA, Ee);
    k_finalize<<<(total + TPB - 1) / TPB, TPB, 0, stream>>>(
        bufB, bufA, dis, b + l * Hd, gamma + l * Hd, beta + l * Hd,
        mean + l * Hd, var + l * Hd, total);
    hin = bufA;
    K = Hd;
  }

  // 3) pooling (gsum | gmax | gcnt are contiguous -> one fill)
  k_fill<<<(2 * Gg * Hd + Gg + TPB - 1) / TPB, TPB, 0, stream>>>(gsum, 2 * Gg * Hd + Gg, 0.0f);
  k_pool<<<(Nn + POOL_CHUNK - 1) / POOL_CHUNK, Hd, 0, stream>>>(bufA, batch, gsum, gmax, gcnt);

  // 4) head MLP
  k_mlp1<<<Gg, Hd, 0, stream>>>(gsum, gmax, gcnt, poolW, poolB, hid);
  k_mlp2<<<Gg, Hd / 2, 0, stream>>>(hid, outW, outB, out);
}